// imgs4dto3d_764504178714
// MI455X (gfx1250) — compile-verified
//
#include <hip/hip_runtime.h>
#include <stdint.h>

// Problem constants (match reference)
constexpr int B_    = 64;
constexpr int E_    = 512;
constexpr int P_    = 31;
constexpr int PP_   = P_ * P_;          // 961 floats / patch, contiguous in memory
constexpr int S_    = 200;
constexpr int CAN_  = S_ * S_;          // 40000 floats = 160000 B canvas
constexpr int HALF_ = 15;

// Tiling: 4 blocks per batch, 16 wave32 per block, one patch per wave per group.
constexpr int EPB_     = 128;           // emitters per block
constexpr int WAVES_   = 16;
constexpr int THREADS_ = WAVES_ * 32;   // 512
constexpr int GRP_     = WAVES_;        // 16 patches staged per group
constexpr int NGRP_    = EPB_ / GRP_;   // 8 groups
constexpr int SLOT_    = 3856;          // patch slot bytes (3844 padded to 16B)
constexpr int ISSUES_  = 9;             // async instrs per wave per patch: 8 x b128 + 1 x b32
constexpr size_t LDS_BYTES_ =
    (size_t)CAN_ * 4 + 2ull * GRP_ * SLOT_;   // 160000 + 123392 = 283392 <= 320KB

__global__ void zero_out_kernel(float* __restrict__ out, int n) {
    int i = blockIdx.x * blockDim.x + threadIdx.x;
    if (i < n) out[i] = 0.0f;
}

__global__ __launch_bounds__(THREADS_) void scatter_patches_kernel(
    const float* __restrict__ imgs,   // [B,E,31,31] f32
    const int*   __restrict__ xyz,    // [B,E,3] i32
    float*       __restrict__ out)    // [B,1,200,200] f32, pre-zeroed
{
    extern __shared__ float smem[];
    float* canvas = smem;                    // CAN_ floats
    float* stagef = smem + CAN_;             // 2 * GRP_ * SLOT_/4 floats

    const int tid  = threadIdx.x;
    const int wave = tid >> 5;               // 0..15, owns one patch per group
    const int lane = tid & 31;
    const int b    = blockIdx.x >> 2;        // batch
    const int e0   = (blockIdx.x & 3) * EPB_;

    // Zero LDS canvas (all waves cooperate)
    for (int i = tid; i < CAN_; i += THREADS_) canvas[i] = 0.0f;

    // Low 32 bits of a generic LDS pointer == LDS byte offset (aperture layout)
    const uint32_t stagingLds = (uint32_t)(uintptr_t)(smem + CAN_);

    // Wave-private async stage of this wave's patch of group `grp` into buffer `buf`.
    // Exactly ISSUES_ (= 9) async instructions per call per wave.
    auto stage = [&](int grp, int buf) {
        const int      e    = e0 + grp * GRP_ + wave;
        const uint32_t lds  = stagingLds + (uint32_t)(buf * GRP_ + wave) * SLOT_;
        const uint32_t goff = (uint32_t)(b * E_ + e) * (uint32_t)(PP_ * 4);
        for (int k = lane; k < 240; k += 32) {           // 8 wave-level issues
            asm volatile("global_load_async_to_lds_b128 %0, %1, %2"
                         :: "v"(lds + (uint32_t)k * 16u),
                            "v"(goff + (uint32_t)k * 16u),
                            "s"(imgs) : "memory");
        }
        if (lane == 0) {                                  // final 4B (3840..3844)
            asm volatile("global_load_async_to_lds_b32 %0, %1, %2"
                         :: "v"(lds + 3840u), "v"(goff + 3840u),
                            "s"(imgs) : "memory");
        }
    };

    // Wave-private accumulate of this wave's staged patch: lane = column,
    // loop = rows. No divisions; conflict-free LDS reads; consecutive-bank atomics.
    auto consume = [&](int grp, int buf) {
        const int e     = e0 + grp * GRP_ + wave;
        const int base3 = (b * E_ + e) * 3;
        const int x = xyz[base3 + 0];                    // row center
        const int y = xyz[base3 + 1];                    // col center
        const float* ps = stagef + (buf * GRP_ + wave) * (SLOT_ / 4);
        if (lane < P_) {
            float* crow = canvas + (x - HALF_) * S_ + (y - HALF_ + lane);
            #pragma unroll
            for (int r = 0; r < P_; ++r) {
                atomicAdd(crow + r * S_, ps[r * P_ + lane]);   // ds_add_f32
            }
        }
    };

    // Prologue: stage group 0; barrier covers canvas zeroing (no wait needed yet).
    stage(0, 0);
    __syncthreads();

    // Wave-private double-buffered pipeline, no block barriers:
    // async loads retire in order, so asynccnt <= ISSUES_ proves the previous
    // patch landed in LDS while the next patch's 9 transfers remain in flight.
    for (int grp = 0; grp < NGRP_; ++grp) {
        if (grp + 1 < NGRP_) {
            stage(grp + 1, (grp + 1) & 1);
            asm volatile("s_wait_asynccnt %0" :: "i"(ISSUES_) : "memory");
        } else {
            asm volatile("s_wait_asynccnt 0x0" ::: "memory");
        }
        consume(grp, grp & 1);
    }

    // All waves' LDS atomics must land before the flush reads the canvas.
    __syncthreads();

    // Flush: 4 blocks per batch combine via hardware fp32 global atomics.
    float* dst = out + b * CAN_;
    for (int i = tid; i < CAN_; i += THREADS_) {
        unsafeAtomicAdd(&dst[i], canvas[i]);             // global_atomic_add_f32
    }
}

extern "C" void kernel_launch(void* const* d_in, const int* in_sizes, int n_in,
                              void* d_out, int out_size, void* d_ws, size_t ws_size,
                              hipStream_t stream) {
    const float* imgs = (const float*)d_in[0];   // [B,E,31,31] f32
    const int*   xyz  = (const int*)d_in[1];     // [B,E,3] i32
    float*       out  = (float*)d_out;           // [B,1,200,200] f32

    const int n = B_ * CAN_;
    zero_out_kernel<<<(n + 255) / 256, 256, 0, stream>>>(out, n);

    const int blocks = B_ * (E_ / EPB_);         // 256
    scatter_patches_kernel<<<blocks, THREADS_, LDS_BYTES_, stream>>>(imgs, xyz, out);
}